// Bottleneck_67525475828280
// MI455X (gfx1250) — compile-verified
//
#include <hip/hip_runtime.h>
#include <cstdint>

#define EPSF   1e-6f
#define BNEPS  1e-5f
#define TEMPI  2.0f      /* 1/TEMP, TEMP=0.5 */
#define NSAMP  5
#define RAD    5
#define LPITCH 20        /* LDS row pitch (floats): 80B rows -> 16B-aligned, conflict-free */

typedef __attribute__((ext_vector_type(2))) float v2f;
typedef __attribute__((ext_vector_type(8))) float v8f;

// ---- CDNA5 async global->LDS helpers (ASYNCcnt-tracked, bypass VGPRs) ----------
__device__ __forceinline__ void async_ld_b32(const void* gp, void* lp) {
  asm volatile("global_load_async_to_lds_b32 %0, %1, off"
               :: "v"((unsigned)(uintptr_t)lp), "v"((unsigned long long)(uintptr_t)gp)
               : "memory");
}
__device__ __forceinline__ void async_ld_b128(const void* gp, void* lp) {
  asm volatile("global_load_async_to_lds_b128 %0, %1, off"
               :: "v"((unsigned)(uintptr_t)lp), "v"((unsigned long long)(uintptr_t)gp)
               : "memory");
}
__device__ __forceinline__ void wait_async0() {
  asm volatile("s_wait_asynccnt 0" ::: "memory");
}

// ---------------- deterministic counter-based RNG (threefry2x32-style) ----------
__device__ __forceinline__ uint32_t rotl32(uint32_t x, int r) { return (x << r) | (x >> (32 - r)); }

__device__ inline uint32_t tf2x32(uint32_t k0, uint32_t k1, uint32_t c0, uint32_t c1) {
  uint32_t ks2 = 0x1BD11BDAu ^ k0 ^ k1;
  uint32_t x0 = c0 + k0, x1 = c1 + k1;
  const int RA[4] = {13, 15, 26, 6};
  const int RB[4] = {17, 29, 16, 24};
#pragma unroll
  for (int i = 0; i < 4; ++i) { x0 += x1; x1 = rotl32(x1, RA[i]); x1 ^= x0; }
  x0 += k1; x1 += ks2 + 1u;
#pragma unroll
  for (int i = 0; i < 4; ++i) { x0 += x1; x1 = rotl32(x1, RB[i]); x1 ^= x0; }
  x0 += ks2; x1 += k0 + 2u;
#pragma unroll
  for (int i = 0; i < 4; ++i) { x0 += x1; x1 = rotl32(x1, RA[i]); x1 ^= x0; }
  x0 += k0; x1 += k1 + 3u;
#pragma unroll
  for (int i = 0; i < 4; ++i) { x0 += x1; x1 = rotl32(x1, RB[i]); x1 ^= x0; }
  x0 += k1; x1 += ks2 + 4u;
#pragma unroll
  for (int i = 0; i < 4; ++i) { x0 += x1; x1 = rotl32(x1, RA[i]); x1 ^= x0; }
  x0 += ks2; x1 += k0 + 5u;
  return x0 ^ x1;
}

__device__ __forceinline__ float gumbel_u32(uint32_t r) {
  float u = ((float)(r >> 8) + 0.5f) * (1.0f / 16777216.0f);  // (0,1)
  return -__logf(-__logf(u));
}

// ---------------- per-out-channel Frobenius norm of W --------------------------
__global__ void k_wnorm(const float* __restrict__ W, float* __restrict__ WN, int K) {
  __shared__ float red[64];
  const int co = blockIdx.x, t = threadIdx.x;
  float s = 0.f;
  for (int i = t; i < K; i += 64) { float v = W[(size_t)co * K + i]; s += v * v; }
  red[t] = s; __syncthreads();
  for (int o = 32; o > 0; o >>= 1) { if (t < o) red[t] += red[t + o]; __syncthreads(); }
  if (t == 0) WN[co] = sqrtf(red[0]);
}

// ---------------- per-position channel sum of squares (-> x_norm) --------------
__global__ void k_sumsq(const float* __restrict__ V, float* __restrict__ S,
                        int C, int HW, int do_sqrt) {
  const int n = blockIdx.x * blockDim.x + threadIdx.x;
  const int b = blockIdx.y;
  if (n >= HW) return;
  const float* Vb = V + (size_t)b * C * HW + n;
  float s = 0.f;
  for (int c = 0; c < C; ++c) { float v = Vb[(size_t)c * HW]; s += v * v; }
  S[(size_t)b * HW + n] = do_sqrt ? sqrtf(s) : s;
}

// 3x3 zero-padded box of channel-sumsq, then sqrt (x_norm for the 3x3 conv)
__global__ void k_box3x3_sqrt(const float* __restrict__ S, float* __restrict__ XN,
                              int HW, int Hi, int Wi) {
  const int n = blockIdx.x * blockDim.x + threadIdx.x;
  const int b = blockIdx.y;
  if (n >= HW) return;
  const int h = n / Wi, w = n % Wi;
  const float* Sb = S + (size_t)b * HW;
  float s = 0.f;
#pragma unroll
  for (int dh = -1; dh <= 1; ++dh)
#pragma unroll
    for (int dw = -1; dw <= 1; ++dw) {
      int h2 = h + dh, w2 = w + dw;
      if ((unsigned)h2 < (unsigned)Hi && (unsigned)w2 < (unsigned)Wi) s += Sb[h2 * Wi + w2];
    }
  XN[(size_t)b * HW + n] = sqrtf(s);
}

// ---------------- 1x1 conv as WMMA GEMM: Y[b,co,n] = sum_c W[co,c] X[b,c,n] ----
// grid (HW/64, Cout/64, B), block 256 (8 waves). Cin % 16 == 0, HW % 64 == 0.
__global__ __launch_bounds__(256)
void k_conv1x1_wmma(const float* __restrict__ X, const float* __restrict__ W,
                    float* __restrict__ Y, int Cin, int Cout, int HW) {
  __shared__ float Wl[64 * LPITCH];   // [row 0..63][k 0..15]
  __shared__ float Xl[64 * LPITCH];   // [n 0..63][k 0..15]
  const int tid = threadIdx.x;
  const int lane = tid & 31, wave = tid >> 5;
  const int lm = lane & 15, hi = lane >> 4;
  const int b = blockIdx.z;
  const int co_base = blockIdx.y * 64;
  const int n_base = blockIdx.x * 64;
  const int nsub = wave & 3;        // 4 n-subtiles of 16
  const int ch = wave >> 2;         // 2 wave groups cover co 0..31 / 32..63
  const float* Xb = X + (size_t)b * Cin * HW;
  float* Yb = Y + (size_t)b * Cout * HW;

  v8f acc0 = {}; v8f acc1 = {};

  const int wr = tid >> 2, wk = (tid & 3) * 4;   // W-staging coords
  const int xn = tid & 63, xk0 = tid >> 6;       // X-staging coords
  const int xcol = n_base + xn;

  for (int kb = 0; kb < Cin; kb += 16) {
    __syncthreads();   // previous iteration's LDS consumers done
    // stage 64x16 W chunk: one async b128 per thread, direct to LDS
    async_ld_b128(W + (size_t)(co_base + wr) * Cin + kb + wk, &Wl[wr * LPITCH + wk]);
    // stage 16x64 X chunk transposed to [n][k]: async b32 gather
#pragma unroll
    for (int j = 0; j < 4; ++j) {
      const int kr = xk0 + j * 4;
      async_ld_b32(Xb + (size_t)(kb + kr) * HW + xcol, &Xl[xn * LPITCH + kr]);
    }
    // warm L2 for the next chunk while WMMA runs
    if (kb + 16 < Cin)
      __builtin_prefetch(Xb + (size_t)(kb + 16 + xk0) * HW + xcol, 0, 2);
    wait_async0();
    __syncthreads();
#pragma unroll
    for (int k0 = 0; k0 < 16; k0 += 4) {
      const int kk = k0 + hi * 2;   // lanes 16..31 supply K+2,K+3
      const v2f bv = *(const v2f*)&Xl[(nsub * 16 + lm) * LPITCH + kk];
      const v2f a0 = *(const v2f*)&Wl[((ch * 2 + 0) * 16 + lm) * LPITCH + kk];
      const v2f a1 = *(const v2f*)&Wl[((ch * 2 + 1) * 16 + lm) * LPITCH + kk];
      acc0 = __builtin_amdgcn_wmma_f32_16x16x4_f32(false, a0, false, bv, (short)0, acc0, false, false);
      acc1 = __builtin_amdgcn_wmma_f32_16x16x4_f32(false, a1, false, bv, (short)0, acc1, false, false);
    }
  }
  const int col = n_base + nsub * 16 + lm;
  const int r0 = co_base + (ch * 2 + 0) * 16 + hi * 8;
  const int r1 = co_base + (ch * 2 + 1) * 16 + hi * 8;
#pragma unroll
  for (int r = 0; r < 8; ++r) {
    Yb[(size_t)(r0 + r) * HW + col] = acc0[r];
    Yb[(size_t)(r1 + r) * HW + col] = acc1[r];
  }
}

// ---------------- 3x3 conv (64->64, pad 1) as 9 tap-GEMMs via WMMA -------------
__global__ __launch_bounds__(256)
void k_conv3x3_wmma(const float* __restrict__ X, const float* __restrict__ W,
                    float* __restrict__ Y, int HW, int Hi, int Wi) {
  __shared__ float Wl[64 * LPITCH];
  __shared__ float Xl[64 * LPITCH];
  const int tid = threadIdx.x;
  const int lane = tid & 31, wave = tid >> 5;
  const int lm = lane & 15, hi = lane >> 4;
  const int b = blockIdx.z;
  const int n_base = blockIdx.x * 64;
  const int nsub = wave & 3, ch = wave >> 2;
  const int C = 64;
  const float* Xb = X + (size_t)b * C * HW;
  float* Yb = Y + (size_t)b * C * HW;

  v8f acc0 = {}; v8f acc1 = {};

  const int wr = tid >> 2, wk0 = (tid & 3) * 4;
  const int xn = tid & 63, xk0 = tid >> 6;
  const int ng = n_base + xn;
  const int hh = ng / Wi, ww = ng % Wi;

  for (int tap = 0; tap < 9; ++tap) {
    const int dh = tap / 3 - 1, dw = tap % 3 - 1;
    const int h2 = hh + dh, w2 = ww + dw;
    const bool ok = ((unsigned)h2 < (unsigned)Hi) && ((unsigned)w2 < (unsigned)Wi);
    const int src = h2 * Wi + w2;
    for (int kb = 0; kb < C; kb += 16) {
      __syncthreads();
      // W[co][c][kh][kw]: per-tap element stride 9 -> async b32 gather to LDS
#pragma unroll
      for (int j = 0; j < 4; ++j)
        async_ld_b32(W + ((size_t)wr * 64 + kb + wk0 + j) * 9 + tap,
                     &Wl[wr * LPITCH + wk0 + j]);
      // shifted, zero-padded B chunk
      if (ok) {
#pragma unroll
        for (int j = 0; j < 4; ++j) {
          const int kr = xk0 + j * 4;
          async_ld_b32(Xb + (size_t)(kb + kr) * HW + src, &Xl[xn * LPITCH + kr]);
        }
      } else {
#pragma unroll
        for (int j = 0; j < 4; ++j) Xl[xn * LPITCH + xk0 + j * 4] = 0.0f;
      }
      wait_async0();
      __syncthreads();
#pragma unroll
      for (int k0 = 0; k0 < 16; k0 += 4) {
        const int kk = k0 + hi * 2;
        const v2f bv = *(const v2f*)&Xl[(nsub * 16 + lm) * LPITCH + kk];
        const v2f a0 = *(const v2f*)&Wl[((ch * 2 + 0) * 16 + lm) * LPITCH + kk];
        const v2f a1 = *(const v2f*)&Wl[((ch * 2 + 1) * 16 + lm) * LPITCH + kk];
        acc0 = __builtin_amdgcn_wmma_f32_16x16x4_f32(false, a0, false, bv, (short)0, acc0, false, false);
        acc1 = __builtin_amdgcn_wmma_f32_16x16x4_f32(false, a1, false, bv, (short)0, acc1, false, false);
      }
    }
  }
  const int col = n_base + nsub * 16 + lm;
  const int r0 = (ch * 2 + 0) * 16 + hi * 8;
  const int r1 = (ch * 2 + 1) * 16 + hi * 8;
#pragma unroll
  for (int r = 0; r < 8; ++r) {
    Yb[(size_t)(r0 + r) * HW + col] = acc0[r];
    Yb[(size_t)(r1 + r) * HW + col] = acc1[r];
  }
}

// ---------------- separable 11x11 box over cosine (cosine computed on the fly) --
__global__ void k_box_row(const float* __restrict__ Y, const float* __restrict__ WN,
                          const float* __restrict__ XN, float* __restrict__ T,
                          int C, int HW, int Wi, long total) {
  const long i = (long)blockIdx.x * blockDim.x + threadIdx.x;
  if (i >= total) return;
  const int n = (int)(i % HW);
  const long bc = i / HW;
  const int c = (int)(bc % C);
  const int b = (int)(bc / C);
  const int h = n / Wi, w = n % Wi;
  const float invw = 1.0f / (WN[c] + EPSF);
  const float* Yr = Y + bc * HW + (size_t)h * Wi;
  const float* Xr = XN + (size_t)b * HW + (size_t)h * Wi;
  float s = 0.f;
#pragma unroll
  for (int d = -RAD; d <= RAD; ++d) {
    const int w2 = w + d;
    if ((unsigned)w2 < (unsigned)Wi) s += Yr[w2] * invw / (Xr[w2] + EPSF);
  }
  T[i] = s;
}

__global__ void k_box_col(const float* __restrict__ T, float* __restrict__ CM,
                          int HW, int Wi, int Hi, long total) {
  const long i = (long)blockIdx.x * blockDim.x + threadIdx.x;
  if (i >= total) return;
  const int n = (int)(i % HW);
  const long base = i - n;
  const int h = n / Wi, w = n % Wi;
  float s = 0.f;
#pragma unroll
  for (int d = -RAD; d <= RAD; ++d) {
    const int h2 = h + d;
    if ((unsigned)h2 < (unsigned)Hi) s += T[base + h2 * Wi + w];
  }
  CM[i] = s * (1.0f / 121.0f);
}

// ---------------- Gumbel-max multinomial + activation term ----------------------
__global__ __launch_bounds__(256)
void k_sample_act(const float* __restrict__ Y, const float* __restrict__ WN,
                  const float* __restrict__ XN, const float* __restrict__ CM,
                  float* __restrict__ acts, int C, int HW, uint32_t seed) {
  __shared__ float sv[NSAMP][256];
  __shared__ int   si[NSAMP][256];
  const int bc = blockIdx.x, t = threadIdx.x;
  const int c = bc % C, b = bc / C;
  const float* Yr = Y + (size_t)bc * HW;
  const float* Xr = XN + (size_t)b * HW;
  const float invw = 1.0f / (WN[c] + EPSF);
  float best[NSAMP]; int bix[NSAMP];
#pragma unroll
  for (int s = 0; s < NSAMP; ++s) { best[s] = -3.4e38f; bix[s] = 0; }
  for (int n = t; n < HW; n += 256) {
    const float logit = (Yr[n] * invw / (Xr[n] + EPSF)) * TEMPI;
#pragma unroll
    for (int s = 0; s < NSAMP; ++s) {
      const uint32_t r = tf2x32(seed, 0x85EBCA6Bu * (uint32_t)(s + 1), (uint32_t)bc, (uint32_t)n);
      const float v = logit + gumbel_u32(r);
      if (v > best[s]) { best[s] = v; bix[s] = n; }
    }
  }
#pragma unroll
  for (int s = 0; s < NSAMP; ++s) { sv[s][t] = best[s]; si[s][t] = bix[s]; }
  for (int o = 128; o > 0; o >>= 1) {
    __syncthreads();
    if (t < o) {
#pragma unroll
      for (int s = 0; s < NSAMP; ++s) {
        const float v2 = sv[s][t + o]; const int i2 = si[s][t + o];
        if (v2 > sv[s][t] || (v2 == sv[s][t] && i2 < si[s][t])) { sv[s][t] = v2; si[s][t] = i2; }
      }
    }
  }
  __syncthreads();
  if (t == 0) {
    float a = 0.f;
#pragma unroll
    for (int s = 0; s < NSAMP; ++s) a += CM[(size_t)bc * HW + si[s][0]];
    acts[bc] = a * (1.0f / NSAMP);
  }
}

__global__ void k_mean(const float* __restrict__ V, int n, float* __restrict__ out) {
  __shared__ float red[256];
  const int t = threadIdx.x;
  float s = 0.f;
  for (int i = t; i < n; i += 256) s += V[i];
  red[t] = s; __syncthreads();
  for (int o = 128; o > 0; o >>= 1) { if (t < o) red[t] += red[t + o]; __syncthreads(); }
  if (t == 0) out[0] = red[0] / (float)n;
}

// ---------------- BatchNorm (training batch stats) + ReLU -----------------------
__global__ void k_bn_stats(const float* __restrict__ Y, float* __restrict__ M,
                           float* __restrict__ R, int C, int HW, int B) {
  __shared__ float s1[256], s2[256];
  const int c = blockIdx.x, t = threadIdx.x;
  const long per = (long)B * HW;
  float a = 0.f, q = 0.f;
  for (long i = t; i < per; i += 256) {
    const int b = (int)(i / HW), n = (int)(i % HW);
    const float v = Y[((size_t)b * C + c) * HW + n];
    a += v; q += v * v;
  }
  s1[t] = a; s2[t] = q; __syncthreads();
  for (int o = 128; o > 0; o >>= 1) {
    if (t < o) { s1[t] += s1[t + o]; s2[t] += s2[t + o]; }
    __syncthreads();
  }
  if (t == 0) {
    const float m = s1[0] / (float)per;
    const float var = s2[0] / (float)per - m * m;
    M[c] = m; R[c] = rsqrtf(var + BNEPS);
  }
}

__global__ void k_bn_relu(const float* __restrict__ Y, float* __restrict__ O,
                          const float* __restrict__ M, const float* __restrict__ R,
                          const float* __restrict__ G, const float* __restrict__ Bb,
                          int C, int HW, long total) {
  const long i = (long)blockIdx.x * blockDim.x + threadIdx.x;
  if (i >= total) return;
  const int c = (int)((i / HW) % C);
  O[i] = fmaxf((Y[i] - M[c]) * R[c] * G[c] + Bb[c], 0.f);
}

__global__ void k_bn_add_relu(const float* __restrict__ Y, const float* __restrict__ Xin,
                              float* __restrict__ O, const float* __restrict__ M,
                              const float* __restrict__ R, const float* __restrict__ G,
                              const float* __restrict__ Bb, int C, int HW, long total) {
  const long i = (long)blockIdx.x * blockDim.x + threadIdx.x;
  if (i >= total) return;
  const int c = (int)((i / HW) % C);
  O[i] = fmaxf((Y[i] - M[c]) * R[c] * G[c] + Bb[c] + Xin[i], 0.f);
}

__global__ void k_acc3(const float* __restrict__ a, float* __restrict__ out) {
  if (threadIdx.x == 0 && blockIdx.x == 0) out[0] = a[0] + a[1] + a[2];
}

// -------------------------------------------------------------------------------
extern "C" void kernel_launch(void* const* d_in, const int* in_sizes, int n_in,
                              void* d_out, int out_size, void* d_ws, size_t ws_size,
                              hipStream_t stream) {
  const int B = 16, Hi = 56, Wi = 56, HW = Hi * Wi;       // 3136 = 49*64
  const long S1 = (long)B * 64 * HW;                      // 3,211,264
  const long S3 = (long)B * 256 * HW;                     // 12,845,056
  const long HWB = (long)B * HW;

  const float* x  = (const float*)d_in[0];
  const float* w1 = (const float*)d_in[1];
  const float* w2 = (const float*)d_in[2];
  const float* w3 = (const float*)d_in[3];
  const float* g1 = (const float*)d_in[4];
  const float* b1 = (const float*)d_in[5];
  const float* g2 = (const float*)d_in[6];
  const float* b2 = (const float*)d_in[7];
  const float* g3 = (const float*)d_in[8];
  const float* b3 = (const float*)d_in[9];
  float* out = (float*)d_out;

  float* ws = (float*)d_ws;
  size_t o = 0;
  auto alloc = [&](long n) { float* p = ws + o; o += (size_t)n; return p; };
  float* y1   = alloc(S1);
  float* y2   = alloc(S1);
  float* y3   = alloc(S3);
  float* out1 = alloc(S1);
  float* out2 = alloc(S1);
  float* tmp  = alloc(S3);
  float* cm   = alloc(S3);
  float* xn   = alloc(HWB);
  float* ssq  = alloc(HWB);
  float* wn   = alloc(256);
  float* bnm  = alloc(256);
  float* bnr  = alloc(256);
  float* acts = alloc(4096);
  float* accs = alloc(4);
  (void)ws_size; (void)n_in; (void)in_sizes; (void)out_size;

  const dim3 blk(256);
  const dim3 gSpat((HW + 255) / 256, B);
  const long g1tot = S1, g3tot = S3;
  const dim3 gE1((unsigned)((S1 + 255) / 256)), gE3((unsigned)((S3 + 255) / 256));

  // ===== stage 1: 1x1 conv 256->64 =====
  k_wnorm<<<64, 64, 0, stream>>>(w1, wn, 256);
  k_sumsq<<<gSpat, blk, 0, stream>>>(x, xn, 256, HW, 1);
  k_conv1x1_wmma<<<dim3(HW / 64, 1, B), blk, 0, stream>>>(x, w1, y1, 256, 64, HW);
  k_box_row<<<gE1, blk, 0, stream>>>(y1, wn, xn, tmp, 64, HW, Wi, g1tot);
  k_box_col<<<gE1, blk, 0, stream>>>(tmp, cm, HW, Wi, Hi, g1tot);
  k_sample_act<<<B * 64, blk, 0, stream>>>(y1, wn, xn, cm, acts, 64, HW, 0x2A2A2A01u);
  k_mean<<<1, blk, 0, stream>>>(acts, B * 64, accs + 0);
  k_bn_stats<<<64, blk, 0, stream>>>(y1, bnm, bnr, 64, HW, B);
  k_bn_relu<<<gE1, blk, 0, stream>>>(y1, out1, bnm, bnr, g1, b1, 64, HW, g1tot);

  // ===== stage 2: 3x3 conv 64->64, pad 1 =====
  k_wnorm<<<64, 64, 0, stream>>>(w2, wn, 64 * 9);
  k_sumsq<<<gSpat, blk, 0, stream>>>(out1, ssq, 64, HW, 0);
  k_box3x3_sqrt<<<gSpat, blk, 0, stream>>>(ssq, xn, HW, Hi, Wi);
  k_conv3x3_wmma<<<dim3(HW / 64, 1, B), blk, 0, stream>>>(out1, w2, y2, HW, Hi, Wi);
  k_box_row<<<gE1, blk, 0, stream>>>(y2, wn, xn, tmp, 64, HW, Wi, g1tot);
  k_box_col<<<gE1, blk, 0, stream>>>(tmp, cm, HW, Wi, Hi, g1tot);
  k_sample_act<<<B * 64, blk, 0, stream>>>(y2, wn, xn, cm, acts, 64, HW, 0x2A2A2A02u);
  k_mean<<<1, blk, 0, stream>>>(acts, B * 64, accs + 1);
  k_bn_stats<<<64, blk, 0, stream>>>(y2, bnm, bnr, 64, HW, B);
  k_bn_relu<<<gE1, blk, 0, stream>>>(y2, out2, bnm, bnr, g2, b2, 64, HW, g1tot);

  // ===== stage 3: 1x1 conv 64->256 + residual =====
  k_wnorm<<<256, 64, 0, stream>>>(w3, wn, 64);
  k_sumsq<<<gSpat, blk, 0, stream>>>(out2, xn, 64, HW, 1);
  k_conv1x1_wmma<<<dim3(HW / 64, 4, B), blk, 0, stream>>>(out2, w3, y3, 64, 256, HW);
  k_box_row<<<gE3, blk, 0, stream>>>(y3, wn, xn, tmp, 256, HW, Wi, g3tot);
  k_box_col<<<gE3, blk, 0, stream>>>(tmp, cm, HW, Wi, Hi, g3tot);
  k_sample_act<<<B * 256, blk, 0, stream>>>(y3, wn, xn, cm, acts, 256, HW, 0x2A2A2A03u);
  k_mean<<<1, blk, 0, stream>>>(acts, B * 256, accs + 2);
  k_bn_stats<<<256, blk, 0, stream>>>(y3, bnm, bnr, 256, HW, B);
  k_bn_add_relu<<<gE3, blk, 0, stream>>>(y3, x, out, bnm, bnr, g3, b3, 256, HW, g3tot);

  // acc scalar after the output tensor
  k_acc3<<<1, 32, 0, stream>>>(accs, out + S3);
}